// AttentionLayer_wosubgraph_19267223290402
// MI455X (gfx1250) — compile-verified
//
#include <hip/hip_runtime.h>
#include <hip/hip_bf16.h>
#include <math.h>

// ---------------------------------------------------------------------------
// Problem constants (fixed by the reference)
// ---------------------------------------------------------------------------
#define N_NODES 20000
#define HIDDEN  128
#define NHEADS  8
#define NTYPE   3
#define NCOLS   48            // 24 (U) + 24 (V) projection columns
#define ALPHA   0.2f
#define PARTIAL_BLOCKS 64

typedef float v2f __attribute__((ext_vector_type(2)));
typedef float v8f __attribute__((ext_vector_type(8)));

// ---------------------------------------------------------------------------
// Kernel 1a: partial column sums of feats (for global readout g)
// grid: PARTIAL_BLOCKS x 128 threads; thread d handles column d
// ---------------------------------------------------------------------------
__global__ __launch_bounds__(HIDDEN)
void colsum_partial_kernel(const float* __restrict__ feats,
                           float* __restrict__ partial, int n_nodes) {
    const int d = threadIdx.x;        // column 0..127
    const int b = blockIdx.x;         // 0..PARTIAL_BLOCKS-1
    float s = 0.0f;
    for (int n = b; n < n_nodes; n += PARTIAL_BLOCKS)
        s += feats[n * HIDDEN + d];
    partial[b * HIDDEN + d] = s;
}

// ---------------------------------------------------------------------------
// Kernel 1b: finalize g = mean(feats) and bias[c,h] = g . a[c,h,256:384]
// single block, 128 threads
// ---------------------------------------------------------------------------
__global__ __launch_bounds__(HIDDEN)
void finalize_g_bias_kernel(const float* __restrict__ partial,
                            const float* __restrict__ a,
                            float* __restrict__ bias, int n_nodes) {
    __shared__ float gs[HIDDEN];
    const int d = threadIdx.x;
    float s = 0.0f;
    for (int p = 0; p < PARTIAL_BLOCKS; ++p)
        s += partial[p * HIDDEN + d];
    gs[d] = s / (float)n_nodes;
    __syncthreads();
    if (d < NTYPE * NHEADS) {
        // a layout: [NTYPE][NHEADS][3*HIDDEN]; j = c*8+h -> row j, slice [256:384)
        const float* ap = a + d * (3 * HIDDEN) + 2 * HIDDEN;
        float acc = 0.0f;
        for (int k = 0; k < HIDDEN; ++k)
            acc += gs[k] * ap[k];
        bias[d] = acc;
    }
}

// ---------------------------------------------------------------------------
// Kernel 2: build transposed weight Wt[k][j], j<24: a[j,k]; j>=24: a[j-24,128+k]
// ---------------------------------------------------------------------------
__global__ __launch_bounds__(256)
void build_wt_kernel(const float* __restrict__ a, float* __restrict__ Wt) {
    const int idx = blockIdx.x * 256 + threadIdx.x;
    if (idx >= HIDDEN * NCOLS) return;
    const int k = idx / NCOLS;
    const int j = idx % NCOLS;
    const float v = (j < 24) ? a[j * (3 * HIDDEN) + k]
                             : a[(j - 24) * (3 * HIDDEN) + HIDDEN + k];
    Wt[idx] = v;
}

// ---------------------------------------------------------------------------
// Kernel 3: WMMA GEMM  uv[N_NODES][48] = feats[N_NODES][128] x Wt[128][48]
// One wave32 per block; M-tile = 16 rows; N = 3 tiles of 16; K = 32 steps of 4.
// Uses V_WMMA_F32_16X16X4_F32 (f32 in/out -> matches reference numerics).
// A-frag (16x4 f32): lanes 0-15 -> K={0,1}, lanes 16-31 -> K={2,3}, lane&15 = M.
// B-frag (4x16 f32): lane&15 = N, VGPR0/1 = K pair selected by lane half.
// ---------------------------------------------------------------------------
#define LDS_STRIDE 132   // 128 + 4 pad: lanes hit 64 distinct banks on frag reads

__global__ __launch_bounds__(32)
void gemm_uv_kernel(const float* __restrict__ feats,
                    const float* __restrict__ Wt,
                    float* __restrict__ uv, int n_nodes) {
    __shared__ float tile[16 * LDS_STRIDE];
    const int tileM = blockIdx.x * 16;
    const int lane  = threadIdx.x;        // 0..31
    const int hi    = lane >> 4;          // 0: K-pair {0,1}, 1: K-pair {2,3}
    const int lm    = lane & 15;          // M (for A) / N-local (for B, store)

    // Stage 16x128 feats tile into LDS (coalesced float4 loads).
    for (int i = lane; i < 16 * (HIDDEN / 4); i += 32) {
        const int r  = i >> 5;            // row 0..15
        const int c4 = (i & 31) << 2;     // col 0,4,...,124
        int row = tileM + r;
        if (row >= n_nodes) row = n_nodes - 1;   // clamp (stores are guarded)
        const float4 v = *reinterpret_cast<const float4*>(feats + row * HIDDEN + c4);
        *reinterpret_cast<float4*>(&tile[r * LDS_STRIDE + c4]) = v;
    }
    __syncthreads();

    v8f acc0 = {}; v8f acc1 = {}; v8f acc2 = {};

    for (int kk = 0; kk < HIDDEN / 4; ++kk) {
        const int k0 = kk * 4 + hi * 2;   // this lane-half's K pair
        v2f afrag;
        afrag.x = tile[lm * LDS_STRIDE + k0];
        afrag.y = tile[lm * LDS_STRIDE + k0 + 1];

        const float* w0 = Wt + k0 * NCOLS;        // row k0
        const float* w1 = w0 + NCOLS;             // row k0+1

        v2f b;
        b.x = w0[lm];       b.y = w1[lm];
        acc0 = __builtin_amdgcn_wmma_f32_16x16x4_f32(false, afrag, false, b,
                                                     (short)0, acc0, false, false);
        b.x = w0[16 + lm];  b.y = w1[16 + lm];
        acc1 = __builtin_amdgcn_wmma_f32_16x16x4_f32(false, afrag, false, b,
                                                     (short)0, acc1, false, false);
        b.x = w0[32 + lm];  b.y = w1[32 + lm];
        acc2 = __builtin_amdgcn_wmma_f32_16x16x4_f32(false, afrag, false, b,
                                                     (short)0, acc2, false, false);
    }

    // C/D layout: VGPR v holds M = v (lanes 0-15) or v+8 (lanes 16-31), N = lane&15.
    for (int v = 0; v < 8; ++v) {
        const int m = tileM + (hi ? v + 8 : v);
        if (m < n_nodes) {
            uv[m * NCOLS + lm]      = acc0[v];
            uv[m * NCOLS + 16 + lm] = acc1[v];
            uv[m * NCOLS + 32 + lm] = acc2[v];
        }
    }
}

// ---------------------------------------------------------------------------
// Kernel 4: per-edge scores + LeakyReLU, written straight into d_out
// one thread per (edge, head)
// ---------------------------------------------------------------------------
__global__ __launch_bounds__(256)
void edge_score_kernel(const float* __restrict__ uv,
                       const float* __restrict__ bias,
                       const int* __restrict__ src,
                       const int* __restrict__ nbr,
                       const int* __restrict__ col,
                       float* __restrict__ out, int E) {
    const int tid = blockIdx.x * 256 + threadIdx.x;
    if (tid >= E * NHEADS) return;
    const int e = tid >> 3;
    const int h = tid & 7;
    const int c = col[e];
    const int j = c * NHEADS + h;
    float s = uv[src[e] * NCOLS + j] + uv[nbr[e] * NCOLS + 24 + j] + bias[j];
    s = (s >= 0.0f) ? s : ALPHA * s;
    out[tid] = s;
}

// ---------------------------------------------------------------------------
// Kernel 5: deterministic segmented softmax, in place on d_out.
// seg_ids are sorted/contiguous; the thread at each segment start owns the
// whole segment exclusively (no atomics, bitwise-deterministic).
// ---------------------------------------------------------------------------
__global__ __launch_bounds__(256)
void seg_softmax_kernel(const int* __restrict__ seg,
                        float* __restrict__ out, int E) {
    const int e = blockIdx.x * 256 + threadIdx.x;
    if (e >= E) return;
    const int s = seg[e];
    if (e > 0 && seg[e - 1] == s) return;   // not a segment start
    int end = e + 1;
    while (end < E && seg[end] == s) ++end;

    for (int h = 0; h < NHEADS; ++h) {
        float m = -INFINITY;
        for (int i = e; i < end; ++i)
            m = fmaxf(m, out[i * NHEADS + h]);
        float sum = 0.0f;
        for (int i = e; i < end; ++i) {
            const float ex = expf(out[i * NHEADS + h] - m);
            out[i * NHEADS + h] = ex;
            sum += ex;
        }
        const float inv = 1.0f / sum;
        for (int i = e; i < end; ++i)
            out[i * NHEADS + h] *= inv;
    }
}

// ---------------------------------------------------------------------------
// Launch: inputs in setup_inputs() order:
//   0 feats [20000*128] f32, 1 a [3*8*384] f32,
//   2 src_idx [E] i32, 3 nbr_idx [E] i32, 4 seg_ids [E] i32, 5 col_ids [E] i32
// d_out: [E*8] f32
// ---------------------------------------------------------------------------
extern "C" void kernel_launch(void* const* d_in, const int* in_sizes, int n_in,
                              void* d_out, int out_size, void* d_ws, size_t ws_size,
                              hipStream_t stream) {
    const float* feats = (const float*)d_in[0];
    const float* a     = (const float*)d_in[1];
    const int*   src   = (const int*)d_in[2];
    const int*   nbr   = (const int*)d_in[3];
    const int*   seg   = (const int*)d_in[4];
    const int*   col   = (const int*)d_in[5];
    float*       out   = (float*)d_out;

    const int E       = in_sizes[2];
    const int n_nodes = in_sizes[0] / HIDDEN;

    // Workspace layout (floats)
    float* ws      = (float*)d_ws;
    float* partial = ws;                              // 64*128
    float* bias    = partial + PARTIAL_BLOCKS * HIDDEN; // 32 (padded)
    float* Wt      = bias + 32;                       // 128*48
    float* uv      = Wt + HIDDEN * NCOLS;             // n_nodes*48

    colsum_partial_kernel<<<PARTIAL_BLOCKS, HIDDEN, 0, stream>>>(feats, partial, n_nodes);
    build_wt_kernel<<<(HIDDEN * NCOLS + 255) / 256, 256, 0, stream>>>(a, Wt);
    finalize_g_bias_kernel<<<1, HIDDEN, 0, stream>>>(partial, a, bias, n_nodes);

    const int mtiles = (n_nodes + 15) / 16;
    gemm_uv_kernel<<<mtiles, 32, 0, stream>>>(feats, Wt, uv, n_nodes);

    const int t = E * NHEADS;
    edge_score_kernel<<<(t + 255) / 256, 256, 0, stream>>>(uv, bias, src, nbr, col, out, E);
    seg_softmax_kernel<<<(E + 255) / 256, 256, 0, stream>>>(seg, out, E);
}